// DummyDenoisingModel_54631984005193
// MI455X (gfx1250) — compile-verified
//
#include <hip/hip_runtime.h>
#include <hip/hip_bf16.h>

typedef float v2f __attribute__((ext_vector_type(2)));
typedef float v8f __attribute__((ext_vector_type(8)));

#define D 16
#define NGRAPH 1024

// ---------------------------------------------------------------------------
// Fill a float buffer with zeros (graph-capture safe).
// ---------------------------------------------------------------------------
__global__ void fill_zero_k(float* __restrict__ p, long long n) {
    long long i = (long long)blockIdx.x * blockDim.x + threadIdx.x;
    if (i < n) p[i] = 0.0f;
}

// ---------------------------------------------------------------------------
// out[node][:] = src_rows[ids[node]][:] @ W   (ids == nullptr -> identity)
// M-tiles of 16 nodes per wave; K=16 via 4x V_WMMA_F32_16X16X4_F32 (f32 exact).
// A 16x4 f32 layout: lane l: row=l&15, holds K = (l>>4)*2 .. +1
// B 4x16  f32 layout: lane l: col=l&15, VGPR v holds K = v + (l>>4)*2
// C/D 16x16 f32:      lane l: col=l&15, VGPR r holds row = r + (l>>4)*8
// ---------------------------------------------------------------------------
__global__ void gather_gemm16_k(const float* __restrict__ src_rows,
                                const int* __restrict__ ids,
                                const float* __restrict__ W,
                                float* __restrict__ out,
                                int n_nodes) {
    const int lane  = threadIdx.x & 31;
    const int wave  = (int)((blockIdx.x * blockDim.x + threadIdx.x) >> 5);
    const int nwave = (int)((gridDim.x * blockDim.x) >> 5);
    const int row   = lane & 15;           // M within tile (A side)
    const int col   = lane & 15;           // N (B/C/D side)
    const int koff  = (lane >> 4) << 1;    // 0 or 2
    const int rowb  = (lane >> 4) << 3;    // 0 or 8 (C/D row base)

    // B fragments for the 4 K-chunks (W is [16][16] row-major, B[k][n]=W[k*16+n])
    v2f bb0, bb1, bb2, bb3;
    bb0.x = W[(0  + koff) * D + col]; bb0.y = W[(1  + koff) * D + col];
    bb1.x = W[(4  + koff) * D + col]; bb1.y = W[(5  + koff) * D + col];
    bb2.x = W[(8  + koff) * D + col]; bb2.y = W[(9  + koff) * D + col];
    bb3.x = W[(12 + koff) * D + col]; bb3.y = W[(13 + koff) * D + col];

    const int ntiles = n_nodes >> 4;
    for (int tile = wave; tile < ntiles; tile += nwave) {
        const int node = (tile << 4) + row;
        const int r = ids ? ids[node] : node;
        const float* xrow = src_rows + (size_t)r * D;
        v2f a0 = *(const v2f*)(xrow + 0  + koff);
        v2f a1 = *(const v2f*)(xrow + 4  + koff);
        v2f a2 = *(const v2f*)(xrow + 8  + koff);
        v2f a3 = *(const v2f*)(xrow + 12 + koff);
        v8f c = {};
        c = __builtin_amdgcn_wmma_f32_16x16x4_f32(false, a0, false, bb0, (short)0, c, false, false);
        c = __builtin_amdgcn_wmma_f32_16x16x4_f32(false, a1, false, bb1, (short)0, c, false, false);
        c = __builtin_amdgcn_wmma_f32_16x16x4_f32(false, a2, false, bb2, (short)0, c, false, false);
        c = __builtin_amdgcn_wmma_f32_16x16x4_f32(false, a3, false, bb3, (short)0, c, false, false);
        float* ob = out + ((size_t)(tile << 4) + rowb) * D + col;
        #pragma unroll
        for (int rr = 0; rr < 8; ++rr) ob[(size_t)rr * D] = c[rr];
    }

    // Scalar tail for n_nodes not a multiple of 16 (not hit for N=200000).
    const int tail_start = ntiles << 4;
    if (tail_start < n_nodes) {
        int node = tail_start + (int)(blockIdx.x * blockDim.x + threadIdx.x);
        if (node < n_nodes) {
            const int r = ids ? ids[node] : node;
            const float* xrow = src_rows + (size_t)r * D;
            for (int j = 0; j < D; ++j) {
                float acc = 0.0f;
                for (int k = 0; k < D; ++k) acc += xrow[k] * W[k * D + j];
                out[(size_t)node * D + j] = acc;
            }
        }
    }
}

// ---------------------------------------------------------------------------
// deg[dst[e]] += 1  (fire-and-forget f32 atomics; agg buffer is L2-resident)
// ---------------------------------------------------------------------------
__global__ void degree_k(const int* __restrict__ dst, float* __restrict__ deg,
                         long long E) {
    long long e = (long long)blockIdx.x * blockDim.x + threadIdx.x;
    if (e < E) atomicAdd(deg + dst[e], 1.0f);
}

// dis[i] = rsqrt(deg[i] + 1)  (in place)
__global__ void rsqrt_k(float* __restrict__ d, int n) {
    int i = blockIdx.x * blockDim.x + threadIdx.x;
    if (i < n) d[i] = rsqrtf(d[i] + 1.0f);
}

// ---------------------------------------------------------------------------
// agg[dst] += h[src] * (dis[src]*dis[dst]); 16 lanes per edge (coalesced h[src]
// reads, 16 non-returning atomic f32 adds per edge).
// ---------------------------------------------------------------------------
__global__ void aggregate_k(const float* __restrict__ h,
                            const float* __restrict__ dis,
                            const int* __restrict__ src,
                            const int* __restrict__ dst,
                            float* __restrict__ agg,
                            long long E) {
    long long t = (long long)blockIdx.x * blockDim.x + threadIdx.x;
    long long e = t >> 4;
    int j = (int)(t & 15);
    if (e < E) {
        int s = src[e], d = dst[e];
        float norm = dis[s] * dis[d];
        atomicAdd(agg + (size_t)d * D + j, h[(size_t)s * D + j] * norm);
    }
}

// h[i][j] = relu(agg[i][j] + h[i][j]*dis[i]^2 + bias[j])   (in place on h)
__global__ void node_relu_k(float* __restrict__ h, const float* __restrict__ agg,
                            const float* __restrict__ dis,
                            const float* __restrict__ bias, long long n16) {
    long long t = (long long)blockIdx.x * blockDim.x + threadIdx.x;
    if (t < n16) {
        long long i = t >> 4; int j = (int)(t & 15);
        float di = dis[i];
        float v = agg[t] + h[t] * (di * di) + bias[j];
        h[t] = v > 0.0f ? v : 0.0f;
    }
}

// v = agg2 + h2*dis^2 + b2;  pooled[batch[i]] += v;  cnt[batch[i]] += 1
__global__ void node_pool_k(const float* __restrict__ h2,
                            const float* __restrict__ agg,
                            const float* __restrict__ dis,
                            const float* __restrict__ bias,
                            const int* __restrict__ batch,
                            float* __restrict__ pooled,
                            float* __restrict__ cnt, long long n16) {
    long long t = (long long)blockIdx.x * blockDim.x + threadIdx.x;
    if (t < n16) {
        long long i = t >> 4; int j = (int)(t & 15);
        float di = dis[i];
        float v = agg[t] + h2[t] * (di * di) + bias[j];
        int g = batch[i];
        atomicAdd(pooled + (size_t)g * D + j, v);
        if (j == 0) atomicAdd(cnt + g, 1.0f);
    }
}

// ---------------------------------------------------------------------------
// out = concat(r_mean, l_mean) @ fcW^T + fcb; split into the two output halves.
// d_out[0 .. 3071]    = out[:, :3] flat
// d_out[3072 .. 6143] = out[:, 3:] flat
// ---------------------------------------------------------------------------
__global__ void fc_k(const float* __restrict__ pooled_r, const float* __restrict__ cnt_r,
                     const float* __restrict__ pooled_l, const float* __restrict__ cnt_l,
                     const float* __restrict__ fcW, const float* __restrict__ fcb,
                     float* __restrict__ out) {
    int t = blockIdx.x * blockDim.x + threadIdx.x;  // g*6 + o
    if (t < NGRAPH * 6) {
        int g = t / 6, o = t - g * 6;
        float icr = 1.0f / fmaxf(cnt_r[g], 1.0f);
        float icl = 1.0f / fmaxf(cnt_l[g], 1.0f);
        float acc = fcb[o];
        #pragma unroll
        for (int j = 0; j < D; ++j)
            acc += fcW[o * (2 * D) + j] * (pooled_r[g * D + j] * icr);
        #pragma unroll
        for (int j = 0; j < D; ++j)
            acc += fcW[o * (2 * D) + D + j] * (pooled_l[g * D + j] * icl);
        if (o < 3) out[g * 3 + o] = acc;
        else       out[NGRAPH * 3 + g * 3 + (o - 3)] = acc;
    }
}

// ---------------------------------------------------------------------------
static void run_protein(const float* emb, const float* W1, const float* b1,
                        const float* W2, const float* b2,
                        const int* ids, const int* edge, const int* batch,
                        int N, long long E,
                        float* bufA, float* bufB, float* dis,
                        float* pooled, float* cnt, hipStream_t stream) {
    const int* src = edge;
    const int* dst = edge + E;
    const long long n16 = (long long)N * D;
    const int B = 256;
    dim3 blk(B);
    int gemm_grid = (int)(((N >> 4) + 7) / 8);           // 8 waves / block
    if (gemm_grid < 1) gemm_grid = 1;

    fill_zero_k<<<(int)((N + B - 1) / B), blk, 0, stream>>>(dis, N);
    fill_zero_k<<<(int)((n16 + B - 1) / B), blk, 0, stream>>>(bufB, n16);

    // h1 = emb[ids] @ W1
    gather_gemm16_k<<<gemm_grid, blk, 0, stream>>>(emb, ids, W1, bufA, N);

    degree_k<<<(int)((E + B - 1) / B), blk, 0, stream>>>(dst, dis, E);
    rsqrt_k<<<(int)((N + B - 1) / B), blk, 0, stream>>>(dis, N);

    aggregate_k<<<(int)((E * 16 + B - 1) / B), blk, 0, stream>>>(bufA, dis, src, dst, bufB, E);
    node_relu_k<<<(int)((n16 + B - 1) / B), blk, 0, stream>>>(bufA, bufB, dis, b1, n16);

    // h2 = relu(h1) @ W2  (identity gather; fully overwrites bufB)
    gather_gemm16_k<<<gemm_grid, blk, 0, stream>>>(bufA, nullptr, W2, bufB, N);

    fill_zero_k<<<(int)((n16 + B - 1) / B), blk, 0, stream>>>(bufA, n16);
    aggregate_k<<<(int)((E * 16 + B - 1) / B), blk, 0, stream>>>(bufB, dis, src, dst, bufA, E);

    fill_zero_k<<<(int)((NGRAPH * D + B - 1) / B), blk, 0, stream>>>(pooled, NGRAPH * D);
    fill_zero_k<<<(int)((NGRAPH + B - 1) / B), blk, 0, stream>>>(cnt, NGRAPH);
    node_pool_k<<<(int)((n16 + B - 1) / B), blk, 0, stream>>>(bufB, bufA, dis, b2, batch,
                                                              pooled, cnt, n16);
}

extern "C" void kernel_launch(void* const* d_in, const int* in_sizes, int n_in,
                              void* d_out, int out_size, void* d_ws, size_t ws_size,
                              hipStream_t stream) {
    const float* emb = (const float*)d_in[0];
    const float* W1  = (const float*)d_in[1];
    const float* b1  = (const float*)d_in[2];
    const float* W2  = (const float*)d_in[3];
    const float* b2  = (const float*)d_in[4];
    const float* fcW = (const float*)d_in[5];
    const float* fcb = (const float*)d_in[6];
    const int* rx     = (const int*)d_in[7];
    const int* redge  = (const int*)d_in[8];
    const int* rbatch = (const int*)d_in[9];
    const int* lx     = (const int*)d_in[10];
    const int* ledge  = (const int*)d_in[11];
    const int* lbatch = (const int*)d_in[12];

    const int N = in_sizes[7];                 // 200000
    const long long E = in_sizes[8] / 2;       // 6400000

    char* ws = (char*)d_ws;
    float* bufA = (float*)ws;  ws += (size_t)N * D * sizeof(float);
    float* bufB = (float*)ws;  ws += (size_t)N * D * sizeof(float);
    float* dis  = (float*)ws;  ws += (size_t)N * sizeof(float);
    float* pooled_r = (float*)ws; ws += (size_t)NGRAPH * D * sizeof(float);
    float* pooled_l = (float*)ws; ws += (size_t)NGRAPH * D * sizeof(float);
    float* cnt_r = (float*)ws; ws += (size_t)NGRAPH * sizeof(float);
    float* cnt_l = (float*)ws; ws += (size_t)NGRAPH * sizeof(float);

    run_protein(emb, W1, b1, W2, b2, rx, redge, rbatch, N, E,
                bufA, bufB, dis, pooled_r, cnt_r, stream);
    run_protein(emb, W1, b1, W2, b2, lx, ledge, lbatch, N, E,
                bufA, bufB, dis, pooled_l, cnt_l, stream);

    fc_k<<<(NGRAPH * 6 + 255) / 256, 256, 0, stream>>>(pooled_r, cnt_r, pooled_l, cnt_l,
                                                       fcW, fcb, (float*)d_out);
}